// SoftAttention_75952201662780
// MI455X (gfx1250) — compile-verified
//
#include <hip/hip_runtime.h>
#include <math.h>

typedef __attribute__((ext_vector_type(2))) float v2f;
typedef __attribute__((ext_vector_type(4))) float v4f;
typedef __attribute__((ext_vector_type(8))) float v8f;

#define B_ 64
#define R_ 2048
#define H_ 1024
#define E_ 512
#define D_ 1024

// ---------------------------------------------------------------------------
// Kernel 1: wh[B,E] = hidden[B,H] @ w_h[H,E] using V_WMMA_F32_16X16X4_F32.
// One wave32 per 16x16 output tile; K-loop steps of 4.
// A layout (16x4 f32): lanes 0-15 -> {K=0,K=1} in v0/v1; lanes 16-31 -> {K=2,K=3}.
// B layout (4x16 f32): v0 = row K=0 (lanes 0-15) / K=2 (lanes 16-31); v1 = K=1/K=3.
// C/D layout: VGPR i holds row (i + 8*(lane>=16)), col (lane&15).
// ---------------------------------------------------------------------------
__global__ __launch_bounds__(32) void wh_gemm_wmma(const float* __restrict__ hidden,
                                                   const float* __restrict__ w_h,
                                                   float* __restrict__ wh) {
    const int tile = blockIdx.x;             // 0 .. (B_/16)*(E_/16)-1 = 127
    const int mt   = tile / (E_ / 16);
    const int nt   = tile % (E_ / 16);
    const int m0   = mt * 16;
    const int n0   = nt * 16;
    const int lane = threadIdx.x;            // 0..31
    const int half = lane >> 4;              // 0: K pair {0,1}, 1: K pair {2,3}
    const int l15  = lane & 15;

    v8f c = {};
    for (int k = 0; k < H_; k += 4) {
        v2f a, b;
        const float* arow = hidden + (size_t)(m0 + l15) * H_ + k + half * 2;
        a.x = arow[0];
        a.y = arow[1];
        const float* brow = w_h + (size_t)(k + half * 2) * E_ + n0 + l15;
        b.x = brow[0];
        b.y = brow[E_];
        c = __builtin_amdgcn_wmma_f32_16x16x4_f32(false, a, false, b,
                                                  (short)0, c, false, false);
    }
#pragma unroll
    for (int i = 0; i < 8; ++i) {
        const int row = m0 + i + half * 8;
        const int col = n0 + l15;
        wh[(size_t)row * E_ + col] = c[i];
    }
}

// ---------------------------------------------------------------------------
// Kernel 2: logits[b,r] = sum_e tanh(wh[b,e] + p_att[b,r,e]) * w_alpha[e] + mask[b,r]
// One wave32 per (b,r); float4 loads, shfl_xor wave reduction (wave32).
// ---------------------------------------------------------------------------
__global__ __launch_bounds__(256) void logits_kernel(const float* __restrict__ p_att,
                                                     const float* __restrict__ wh,
                                                     const float* __restrict__ w_alpha,
                                                     const float* __restrict__ mask,
                                                     float* __restrict__ logits) {
    const int gw   = (blockIdx.x * 256 + threadIdx.x) >> 5;  // global wave id
    const int lane = threadIdx.x & 31;
    const int b    = gw / R_;
    const int r    = gw % R_;

    const float* p = p_att + ((size_t)b * R_ + r) * E_;
    const float* w = wh + (size_t)b * E_;

    float acc = 0.0f;
#pragma unroll
    for (int i = 0; i < E_ / 128; ++i) {     // 4 iterations, 128 floats per wave
        const int e = i * 128 + lane * 4;
        v4f pv = *(const v4f*)(p + e);
        v4f wv = *(const v4f*)(w + e);
        v4f av = *(const v4f*)(w_alpha + e);
        acc += tanhf(pv.x + wv.x) * av.x;
        acc += tanhf(pv.y + wv.y) * av.y;
        acc += tanhf(pv.z + wv.z) * av.z;
        acc += tanhf(pv.w + wv.w) * av.w;
    }
#pragma unroll
    for (int off = 16; off > 0; off >>= 1)
        acc += __shfl_xor(acc, off, 32);
    if (lane == 0)
        logits[(size_t)b * R_ + r] = acc + mask[(size_t)b * R_ + r];
}

// ---------------------------------------------------------------------------
// Kernel 3: probs[b,:] = softmax(logits[b,:]) over R. One block per batch row.
// ---------------------------------------------------------------------------
__global__ __launch_bounds__(256) void softmax_kernel(const float* __restrict__ logits,
                                                      float* __restrict__ probs) {
    __shared__ float red[256];
    const int b   = blockIdx.x;
    const int tid = threadIdx.x;
    const float* in = logits + (size_t)b * R_;

    float mx = -INFINITY;
    for (int r = tid; r < R_; r += 256) mx = fmaxf(mx, in[r]);
    red[tid] = mx;
    __syncthreads();
    for (int s = 128; s > 0; s >>= 1) {
        if (tid < s) red[tid] = fmaxf(red[tid], red[tid + s]);
        __syncthreads();
    }
    mx = red[0];
    __syncthreads();

    float sum = 0.0f;
    for (int r = tid; r < R_; r += 256) sum += expf(in[r] - mx);
    red[tid] = sum;
    __syncthreads();
    for (int s = 128; s > 0; s >>= 1) {
        if (tid < s) red[tid] += red[tid + s];
        __syncthreads();
    }
    const float inv = 1.0f / red[0];
    __syncthreads();

    for (int r = tid; r < R_; r += 256)
        probs[(size_t)b * R_ + r] = expf(in[r] - mx) * inv;
}

// ---------------------------------------------------------------------------
// Kernel 4: att[b,d] = sum_r probs[b,r] * att_feats[b,r,d]
// Block = (b, 256-wide d-chunk). probs row staged in LDS (8 KB); att_feats
// streamed exactly once, fully coalesced along d.
// ---------------------------------------------------------------------------
__global__ __launch_bounds__(256) void wsum_kernel(const float* __restrict__ att_feats,
                                                   const float* __restrict__ probs,
                                                   float* __restrict__ out) {
    __shared__ float lp[R_];
    const int b = blockIdx.y;
    const int d = blockIdx.x * 256 + threadIdx.x;

    for (int r = threadIdx.x; r < R_; r += 256)
        lp[r] = probs[(size_t)b * R_ + r];
    __syncthreads();

    const float* af = att_feats + ((size_t)b * R_) * D_ + d;
    float a0 = 0.f, a1 = 0.f, a2 = 0.f, a3 = 0.f;
    for (int r = 0; r < R_; r += 4) {
        a0 += lp[r + 0] * af[(size_t)(r + 0) * D_];
        a1 += lp[r + 1] * af[(size_t)(r + 1) * D_];
        a2 += lp[r + 2] * af[(size_t)(r + 2) * D_];
        a3 += lp[r + 3] * af[(size_t)(r + 3) * D_];
    }
    out[(size_t)b * D_ + d] = (a0 + a1) + (a2 + a3);
}

extern "C" void kernel_launch(void* const* d_in, const int* in_sizes, int n_in,
                              void* d_out, int out_size, void* d_ws, size_t ws_size,
                              hipStream_t stream) {
    const float* hidden   = (const float*)d_in[0];   // [B,H]
    const float* att_f    = (const float*)d_in[1];   // [B,R,D]
    const float* p_att    = (const float*)d_in[2];   // [B,R,E]
    const float* mask     = (const float*)d_in[3];   // [B,R]
    const float* w_h      = (const float*)d_in[4];   // [H,E]
    const float* w_alpha  = (const float*)d_in[5];   // [E]
    float*       out      = (float*)d_out;           // [B,D]

    float* wh     = (float*)d_ws;                    // B*E floats
    float* logits = wh + (size_t)B_ * E_;            // B*R floats
    float* probs  = logits + (size_t)B_ * R_;        // B*R floats

    // 1) wh = hidden @ w_h  (WMMA f32 16x16x4, one wave per tile)
    wh_gemm_wmma<<<dim3((B_ / 16) * (E_ / 16)), dim3(32), 0, stream>>>(hidden, w_h, wh);

    // 2) logits: one wave per (b,r); 8 waves per 256-thread block
    logits_kernel<<<dim3((B_ * R_) / 8), dim3(256), 0, stream>>>(p_att, wh, w_alpha, mask, logits);

    // 3) softmax over regions
    softmax_kernel<<<dim3(B_), dim3(256), 0, stream>>>(logits, probs);

    // 4) weighted sum over regions
    wsum_kernel<<<dim3(D_ / 256, B_), dim3(256), 0, stream>>>(att_f, probs, out);
}